// RelativeBucketedTimeAndPositionBasedBias_5351529251291
// MI455X (gfx1250) — compile-verified
//
#include <hip/hip_runtime.h>
#include <stdint.h>

// ---------------------------------------------------------------------------
// RelativeBucketedTimeAndPositionBasedBias for MI455X (gfx1250)
//
// out[b,i,j] = pos_w[j - i + N - 1]
//            + ts_w[ clip( floor(log2(max(|ts[b,min(i+1,N-1)] - ts[b,j]|, 1))), 0, 128 ) ]
//
// Store-bandwidth-bound (134 MB out @ 23.3 TB/s ~ 6 us). All reads staged
// via CDNA5 async global->LDS (ASYNCcnt) or kept in registers; all stores
// are 128-bit.
// ---------------------------------------------------------------------------

#define BATCH    8
#define SEQ      2048
#define NBUCK    128
#define ROWS     4          // rows (i) per block
#define THREADS  256
#define POS_LDS  2052       // pos_w window: 2048 + (ROWS-1) rounded up to x4

__global__ __launch_bounds__(THREADS)
void rel_bias_kernel(const int*   __restrict__ ts,     // [B, N] int32 timestamps (sorted)
                     const float* __restrict__ ts_w,   // [129]
                     const float* __restrict__ pos_w,  // [2N-1] = [4095]
                     float*       __restrict__ out)    // [B, N, N]
{
    __shared__ __align__(16) float s_pos[POS_LDS];
    __shared__ float s_tsw[NBUCK + 1];

    const int tid  = threadIdx.x;
    const int tile = blockIdx.x;          // 0 .. B*(N/ROWS)-1
    const int b    = tile >> 9;           // N/ROWS = 512 tiles per batch
    const int i0   = (tile & 511) << 2;   // first row of this tile

    // pos_w window needed by this block: indices [2044-i0 .. 4095-i0]
    // start is a multiple of 4 -> 16B-aligned b128 async copies.
    const int   st = (SEQ - ROWS) - i0;   // 2044 - i0
    const float* pw = pos_w + st;

    // ---- async stage pos_w window into LDS (global_load_async_to_lds_b128)
    for (int k = tid; k < POS_LDS / 4; k += THREADS) {
        uint32_t loff = (uint32_t)(uintptr_t)(&s_pos[4 * k]);   // LDS byte addr
        uint64_t ga   = (uint64_t)(uintptr_t)(pw + 4 * k);
        asm volatile("global_load_async_to_lds_b128 %0, %1, off"
                     :: "v"(loff), "v"(ga) : "memory");
    }
    // ---- async stage ts_w[0..128] into LDS (b32 per lane)
    if (tid <= NBUCK) {
        uint32_t loff = (uint32_t)(uintptr_t)(&s_tsw[tid]);
        uint64_t ga   = (uint64_t)(uintptr_t)(ts_w + tid);
        asm volatile("global_load_async_to_lds_b32 %0, %1, off"
                     :: "v"(loff), "v"(ga) : "memory");
    }

    // ---- per-thread timestamp columns: two coalesced b128 loads, reused x4 rows
    const int  j0    = tid * 8;
    const int* tsrow = ts + b * SEQ;
    const int4 c0 = *(const int4*)(tsrow + j0);
    const int4 c1 = *(const int4*)(tsrow + j0 + 4);
    int tsj[8] = {c0.x, c0.y, c0.z, c0.w, c1.x, c1.y, c1.z, c1.w};

    // wait for this wave's async LDS writes, then make them WG-visible
    asm volatile("s_wait_asynccnt 0x0" ::: "memory");
    __syncthreads();

#pragma unroll
    for (int r = 0; r < ROWS; ++r) {
        const int i     = i0 + r;
        const int inext = (i + 1 < SEQ) ? (i + 1) : (SEQ - 1);
        const int tn    = tsrow[inext];          // block-uniform -> scalar load

        float v[8];
#pragma unroll
        for (int c = 0; c < 8; ++c) {
            int      d  = tn - tsj[c];
            unsigned xa = (unsigned)(d < 0 ? -d : d);
            if (xa < 1u) xa = 1u;
            int bkt = 31 - __builtin_clz(xa);    // exact floor(log2(xa))
            if (bkt > NBUCK) bkt = NBUCK;        // clip (upper bound, safety)
            // pos_w[j - i + 2047] == s_pos[j + (ROWS-1) - r]
            v[c] = s_tsw[bkt] + s_pos[j0 + c + (ROWS - 1 - r)];
        }

        float* orow = out + (((size_t)(b * SEQ + i)) << 11) + j0;
        *(float4*)(orow)     = make_float4(v[0], v[1], v[2], v[3]);
        *(float4*)(orow + 4) = make_float4(v[4], v[5], v[6], v[7]);
    }
}

extern "C" void kernel_launch(void* const* d_in, const int* in_sizes, int n_in,
                              void* d_out, int out_size, void* d_ws, size_t ws_size,
                              hipStream_t stream) {
    (void)in_sizes; (void)n_in; (void)out_size; (void)d_ws; (void)ws_size;
    const int*   ts   = (const int*)  d_in[0];   // all_timestamps [8,2048] int32
    const float* tsw  = (const float*)d_in[1];   // ts_w  [129]
    const float* posw = (const float*)d_in[2];   // pos_w [4095]
    float*       out  = (float*)      d_out;     // [8,2048,2048]

    dim3 grid(BATCH * (SEQ / ROWS));             // 4096 blocks
    dim3 block(THREADS);
    rel_bias_kernel<<<grid, block, 0, stream>>>(ts, tsw, posw, out);
}